// getDestroyerLoss_11982958756535
// MI455X (gfx1250) — compile-verified
//
#include <hip/hip_runtime.h>

// ---------------------------------------------------------------------------
// SSIM-blended patch loss for MI455X (gfx1250, wave32, WMMA).
//
// Separable 11-tap Gaussian convolution expressed as banded matmuls on
// V_WMMA_F32_16X16X4_F32.  One workgroup (8 waves) per 64x64 patch.
// ---------------------------------------------------------------------------

typedef __attribute__((ext_vector_type(2))) float v2f;
typedef __attribute__((ext_vector_type(8))) float v8f;

#define PADSTR 76                 // padded input row stride (bank-conflict free)
#define HSTR 76                   // horizontal-result row stride
#define PAD_FLOATS (76 * 76)      // 5776 floats per padded input plane
#define H_FLOATS (76 * 76)        // 5776 floats per horizontal-result plane
#define SMEM_H_OFF (2 * PAD_FLOATS)
#define SMEM_BAND_OFF (2 * PAD_FLOATS + 5 * H_FLOATS)
#define SMEM_FLOATS (SMEM_BAND_OFF + 48)
#define SMEM_BYTES (SMEM_FLOATS * 4)   // 161,920 B -> 2 WGs per 320KB WGP

// Normalized 11-tap Gaussian, sigma = 1.5 (matches reference window).
__constant__ float GK[11] = {
    1.0283797e-3f, 7.5987722e-3f, 3.6000773e-2f, 1.0936069e-1f,
    2.1300553e-1f, 2.6601332e-1f, 2.1300553e-1f, 1.0936069e-1f,
    3.6000773e-2f, 7.5987722e-3f, 1.0283797e-3f};

__device__ __forceinline__ v8f wmma4(v2f a, v2f b, v8f c) {
  // D = A(16x4,f32) * B(4x16,f32) + C(16x16,f32)
  return __builtin_amdgcn_wmma_f32_16x16x4_f32(false, a, false, b, (short)0, c,
                                               false, false);
}

__global__ __launch_bounds__(256) void ssim_patch_loss_kernel(
    const float* __restrict__ A, const float* __restrict__ N,
    const float* __restrict__ GT, const float* __restrict__ G,
    const float* __restrict__ BG, float* __restrict__ partial) {
  extern __shared__ float smem[];
  float* padN = smem;                    // 76x76 zero-padded N patch
  float* padT = smem + PAD_FLOATS;       // 76x76 zero-padded GT patch
  float* hbase = smem + SMEM_H_OFF;      // 5 planes of 76x76 horiz results
  float* band = smem + SMEM_BAND_OFF;    // 48-entry zero-extended gaussian

  const int tid = threadIdx.x;
  const int lane = tid & 31;
  const int wv = tid >> 5;               // wave id 0..7
  const int l = lane & 15;               // low-half lane index
  const int hf = lane >> 4;              // half-wave select

  const int pid = blockIdx.x;            // 0..2047  (b*64 + patch)
  const int b = pid >> 6;
  const int p = pid & 63;
  const int pr = p >> 3, pc = p & 7;
  const long imgBase = (long)b * 786432 + (long)pr * 32768 + (long)pc * 64;

  // ---- init LDS: zero padded planes, build band lookup (band[16+t]=g[t]) ---
  for (int i = tid; i < 2 * PAD_FLOATS; i += 256) smem[i] = 0.0f;
  if (tid < 48) band[tid] = (tid >= 16 && tid <= 26) ? GK[tid - 16] : 0.0f;

  // ---- MSE losses (independent of LDS; coalesced) --------------------------
  float bgAcc = 0.0f, aAcc = 0.0f;
#pragma unroll 4
  for (int i = 0; i < 48; ++i) {
    int idx = tid + (i << 8);            // 0..12287 over (c, ph, pw)
    int c = idx >> 12, rem = idx & 4095;
    int ph = rem >> 6, pw = rem & 63;
    long off = imgBase + (long)c * 262144 + ph * 512 + pw;
    float gv = G[off];
    float d1 = BG[off] - gv; bgAcc += d1 * d1;
    float d2 = A[off] - gv;  aAcc += d2 * d2;
  }

  __syncthreads();  // band table + zeroed halos visible

  // ---- per-lane band fragments (shared by both passes) ---------------------
  // chunk c, element j: value = g[(c*4 + hf*2 + j) - l]  (zero outside band)
  v2f bf[7];
#pragma unroll
  for (int c = 0; c < 7; ++c) {
    int kb = c * 4 + hf * 2;
    bf[c].x = band[kb - l + 16];
    bf[c].y = band[kb + 1 - l + 16];
  }

  float ssimAcc = 0.0f;

  for (int ch = 0; ch < 3; ++ch) {
    const long cbase = imgBase + (long)ch * 262144;

    // ---- stage patch into zero-padded LDS planes ---------------------------
#pragma unroll
    for (int i = 0; i < 16; ++i) {
      int idx = tid + (i << 8);
      int ph = idx >> 6, pw = idx & 63;
      long off = cbase + ph * 512 + pw;
      int pi = (ph + 5) * PADSTR + (pw + 5);
      padN[pi] = N[off];
      padT[pi] = GT[off];
    }
    __syncthreads();

    // ---- horizontal pass: 20 row-block x col-block tiles, 5 quantities -----
    // rows cover padded rows 0..75 via bases {0,16,32,48,60}
    for (int t = wv; t < 20; t += 8) {
      int rb = (t >> 2) * 16; if (rb > 60) rb = 60;
      int cb = (t & 3) << 4;
      v8f aN = {}, aT = {}, aNN = {}, aTT = {}, aNT = {};
#pragma unroll
      for (int c = 0; c < 7; ++c) {
        int pi = (rb + l) * PADSTR + cb + c * 4 + hf * 2;
        float n0 = padN[pi], n1 = padN[pi + 1];
        float t0 = padT[pi], t1 = padT[pi + 1];
        v2f va, bb = bf[c];
        va.x = n0;      va.y = n1;      aN  = wmma4(va, bb, aN);
        va.x = t0;      va.y = t1;      aT  = wmma4(va, bb, aT);
        va.x = n0 * n0; va.y = n1 * n1; aNN = wmma4(va, bb, aNN);
        va.x = t0 * t0; va.y = t1 * t1; aTT = wmma4(va, bb, aTT);
        va.x = n0 * t0; va.y = n1 * t1; aNT = wmma4(va, bb, aNT);
      }
#pragma unroll
      for (int vi = 0; vi < 8; ++vi) {
        int hi = (rb + vi + hf * 8) * HSTR + cb + l;
        hbase[hi]                = aN[vi];
        hbase[H_FLOATS + hi]     = aT[vi];
        hbase[2 * H_FLOATS + hi] = aNN[vi];
        hbase[3 * H_FLOATS + hi] = aTT[vi];
        hbase[4 * H_FLOATS + hi] = aNT[vi];
      }
    }
    __syncthreads();

    // ---- vertical pass + SSIM map: 16 output tiles -------------------------
    for (int t = wv; t < 16; t += 8) {
      int rb = (t >> 2) << 4;            // output row base (patch coords)
      int cb = (t & 3) << 4;
      v8f m1 = {}, m2 = {}, e11 = {}, e22 = {}, e12 = {};
#pragma unroll
      for (int c = 0; c < 7; ++c) {
        v2f aa = bf[c], vb;
        int hi = (rb + c * 4 + hf * 2) * HSTR + cb + l;
        vb.x = hbase[hi];                vb.y = hbase[hi + HSTR];                m1  = wmma4(aa, vb, m1);
        vb.x = hbase[H_FLOATS + hi];     vb.y = hbase[H_FLOATS + hi + HSTR];     m2  = wmma4(aa, vb, m2);
        vb.x = hbase[2 * H_FLOATS + hi]; vb.y = hbase[2 * H_FLOATS + hi + HSTR]; e11 = wmma4(aa, vb, e11);
        vb.x = hbase[3 * H_FLOATS + hi]; vb.y = hbase[3 * H_FLOATS + hi + HSTR]; e22 = wmma4(aa, vb, e22);
        vb.x = hbase[4 * H_FLOATS + hi]; vb.y = hbase[4 * H_FLOATS + hi + HSTR]; e12 = wmma4(aa, vb, e12);
      }
#pragma unroll
      for (int vi = 0; vi < 8; ++vi) {
        float mu1 = m1[vi], mu2 = m2[vi];
        float mu1s = mu1 * mu1, mu2s = mu2 * mu2, mu12 = mu1 * mu2;
        float s1 = e11[vi] - mu1s;
        float s2 = e22[vi] - mu2s;
        float s12 = e12[vi] - mu12;
        float num = (2.0f * mu12 + 1e-4f) * (2.0f * s12 + 9e-4f);
        float den = (mu1s + mu2s + 1e-4f) * (s1 + s2 + 9e-4f);
        ssimAcc += num / den;
      }
    }
    __syncthreads();
  }

  // ---- deterministic block reduction (reuse padN region) -------------------
  float* red = smem;  // 3 x 256 floats
  red[tid] = ssimAcc;
  red[256 + tid] = bgAcc;
  red[512 + tid] = aAcc;
  __syncthreads();
  for (int s = 128; s > 0; s >>= 1) {
    if (tid < s) {
      red[tid] += red[tid + s];
      red[256 + tid] += red[256 + tid + s];
      red[512 + tid] += red[512 + tid + s];
    }
    __syncthreads();
  }
  if (tid == 0) {
    const float inv = 1.0f / 12288.0f;   // mean over (3, 64, 64)
    float ssimMean = red[0] * inv;
    float diff = 1.0f - ssimMean;        // LD = 1.0
    if (diff > 1.0f) diff = 1.0f;        // jnp.minimum(diff, 1)
    float bg = red[256] * inv;
    float aa = red[512] * inv;
    partial[pid] = diff * bg + (1.0f - diff) * aa;
  }
}

__global__ __launch_bounds__(256) void final_reduce_kernel(
    const float* __restrict__ partial, float* __restrict__ out) {
  __shared__ float red[256];
  float s = 0.0f;
  for (int i = threadIdx.x; i < 2048; i += 256) s += partial[i];
  red[threadIdx.x] = s;
  __syncthreads();
  for (int st = 128; st > 0; st >>= 1) {
    if (threadIdx.x < st) red[threadIdx.x] += red[threadIdx.x + st];
    __syncthreads();
  }
  if (threadIdx.x == 0) out[0] = red[0];
}

extern "C" void kernel_launch(void* const* d_in, const int* in_sizes, int n_in,
                              void* d_out, int out_size, void* d_ws,
                              size_t ws_size, hipStream_t stream) {
  (void)in_sizes; (void)n_in; (void)out_size; (void)ws_size;
  const float* A  = (const float*)d_in[0];
  const float* N  = (const float*)d_in[1];
  const float* GT = (const float*)d_in[2];
  const float* G  = (const float*)d_in[3];
  const float* BG = (const float*)d_in[4];
  float* partial = (float*)d_ws;         // 2048 per-patch losses (8 KB)
  float* out = (float*)d_out;

  ssim_patch_loss_kernel<<<dim3(2048), dim3(256), SMEM_BYTES, stream>>>(
      A, N, GT, G, BG, partial);
  final_reduce_kernel<<<dim3(1), dim3(256), 0, stream>>>(partial, out);
}